// GrahamProb_47425028882920
// MI455X (gfx1250) — compile-verified
//
#include <hip/hip_runtime.h>
#include <hip/hip_bf16.h>

typedef __attribute__((ext_vector_type(2))) float v2f;
typedef __attribute__((ext_vector_type(8))) float v8f;

#define CCH   64
#define HH    112
#define WW    112
#define HPW   99            // 112 - 14 + 1
#define PLANE (HH * WW)     // 12544
#define PLANE4 (PLANE / 4)  // 3136
#define PPL   (HPW * HPW)   // 99*99 pooled plane

// Padded row-sum buffer: 128 rows x 112 cols per image; rows >= 112 and
// cols >= 99 are zero so the WMMA pooling kernel can load unconditionally.
#define RROWS 128
#define RCOLS 112
#define RPLP  (RROWS * RCOLS)   // 14336 floats per image

// ---------------------------------------------------------------------------
// Kernel 1: t[b,h,w] = (sum_c feat[b,c,h,w])^2   — the bandwidth-bound stage.
// One thread per 4 consecutive pixels; 64 coalesced b128 loads per thread.
// ---------------------------------------------------------------------------
__global__ void sumsq_kernel(const float* __restrict__ feat,
                             float* __restrict__ t, int total4) {
    int i = blockIdx.x * blockDim.x + threadIdx.x;
    if (i >= total4) return;
    int b = i / PLANE4;
    int p = i - b * PLANE4;
    const float4* base = (const float4*)feat + (size_t)b * CCH * PLANE4 + p;
    float4 s = make_float4(0.f, 0.f, 0.f, 0.f);
#pragma unroll 4
    for (int c = 0; c < CCH; ++c) {
        float4 v = base[(size_t)c * PLANE4];
        if (c + 1 < CCH) __builtin_prefetch(base + (size_t)(c + 1) * PLANE4, 0, 0);
        s.x += v.x; s.y += v.y; s.z += v.z; s.w += v.w;
    }
    float4 o;
    o.x = s.x * s.x; o.y = s.y * s.y; o.z = s.z * s.z; o.w = s.w * s.w;
    ((float4*)t)[(size_t)b * PLANE4 + p] = o;
}

// ---------------------------------------------------------------------------
// Kernel 2: horizontal 14-window sum into the PADDED r buffer:
//   r[b,h,w'] = sum_{k<14} t[b,h,w'+k]  for h<112, w'<99; 0 in the pad region.
// Launched over B*128*112 threads so the pad is rewritten to zero every call.
// ---------------------------------------------------------------------------
__global__ void rowsum_kernel(const float* __restrict__ t,
                              float* __restrict__ r, int total) {
    int i = blockIdx.x * blockDim.x + threadIdx.x;
    if (i >= total) return;
    int b = i / RPLP;
    int rem = i - b * RPLP;
    int h = rem / RCOLS;
    int w = rem - h * RCOLS;
    float s = 0.f;
    if (h < HH && w < HPW) {
        const float* row = t + ((size_t)b * HH + h) * WW + w;
#pragma unroll
        for (int k = 0; k < 14; ++k) s += row[k];
    }
    r[i] = s;
}

// ---------------------------------------------------------------------------
// Kernel 3: vertical 14-window mean via WMMA.
// pooled[h',w'] = (1/196) * sum_k A[h'-h0, k] * r[h0+k, w']  with A a 16x32
// banded ones matrix (A[m,k]=1 iff 0 <= k-m <= 13), done as 8 chained
// V_WMMA_F32_16X16X4_F32 ops. One wave (32 threads) per 16x16 output tile.
// A layout (32-bit A 16x4): lanes=M; VGPR0 holds K=k0 (lo lanes) / k0+2 (hi),
// VGPR1 holds K=k0+1 / k0+3. B layout mirrors with lanes=N.
// r is padded (128x112, zeros outside) so all loads are unconditional.
// ---------------------------------------------------------------------------
__global__ __launch_bounds__(32) void colpool_wmma(const float* __restrict__ r,
                                                   float* __restrict__ pooled) {
    const int tilesPer = 7;                       // ceil(99/16)
    int tile = blockIdx.x;
    int b  = tile / (tilesPer * tilesPer);
    int t2 = tile - b * (tilesPer * tilesPer);
    int h0 = (t2 / tilesPer) * 16;
    int w0 = (t2 % tilesPer) * 16;

    int lane   = threadIdx.x;
    int hiHalf = lane >> 4;        // 0: lanes 0-15, 1: lanes 16-31
    int lol    = lane & 15;

    const float* rb = r + (size_t)b * RPLP + (size_t)h0 * RCOLS + (w0 + lol);
    v8f acc = {};

#pragma unroll
    for (int k0 = 0; k0 < 32; k0 += 4) {
        // A-matrix: banded ones (window selector), per 16x4 f32 A layout
        int kA = k0 + hiHalf * 2;
        int m  = lol;
        v2f a;
        a.x = (kA     >= m && (kA     - m) <= 13) ? 1.0f : 0.0f;
        a.y = (kA + 1 >= m && (kA + 1 - m) <= 13) ? 1.0f : 0.0f;
        // B-matrix: r rows h0+k, columns w0+n — pad guarantees in-bounds
        int hB = k0 + hiHalf * 2;
        v2f bb;
        bb.x = rb[(size_t)hB * RCOLS];
        bb.y = rb[(size_t)(hB + 1) * RCOLS];
        // 8 args: (neg_a, A, neg_b, B, c_mod, C, reuse_a, reuse_b)
        acc = __builtin_amdgcn_wmma_f32_16x16x4_f32(
            false, a, false, bb, (short)0, acc, false, false);
    }

    const float inv = 1.0f / 196.0f;
    float* pb = pooled + (size_t)b * PPL;
#pragma unroll
    for (int j = 0; j < 8; ++j) {
        int row = h0 + j + hiHalf * 8;   // C/D layout: VGPR j -> M=j (lo), M=j+8 (hi)
        int col = w0 + lol;
        if (row < HPW && col < HPW)
            pb[(size_t)row * HPW + col] = acc[j] * inv;
    }
}

// ---------------------------------------------------------------------------
// Kernel 4: sim = C^2 * pooled * (sum of 8 zero-padded neighbors)
// ---------------------------------------------------------------------------
__global__ void neigh_kernel(const float* __restrict__ pooled,
                             float* __restrict__ out, int total) {
    int i = blockIdx.x * blockDim.x + threadIdx.x;
    if (i >= total) return;
    int b = i / PPL;
    int rem = i - b * PPL;
    int h = rem / HPW;
    int w = rem - h * HPW;
    const float* pb = pooled + (size_t)b * PPL;
    float ns = 0.f;
#pragma unroll
    for (int dy = -1; dy <= 1; ++dy)
#pragma unroll
        for (int dx = -1; dx <= 1; ++dx) {
            if (dy == 0 && dx == 0) continue;
            int hh = h + dy, ww = w + dx;
            if (hh >= 0 && hh < HPW && ww >= 0 && ww < HPW)
                ns += pb[(size_t)hh * HPW + ww];
        }
    out[i] = 4096.0f * pb[(size_t)h * HPW + w] * ns;   // C^2 = 64^2
}

// ---------------------------------------------------------------------------
extern "C" void kernel_launch(void* const* d_in, const int* in_sizes, int n_in,
                              void* d_out, int out_size, void* d_ws, size_t ws_size,
                              hipStream_t stream) {
    const float* feat = (const float*)d_in[0];
    float* out = (float*)d_out;

    int B = in_sizes[0] / (CCH * PLANE);   // 32

    float* t      = (float*)d_ws;                          // B * 12544
    float* r      = t + (size_t)B * PLANE;                 // B * 14336 (padded)
    float* pooled = r + (size_t)B * RPLP;                  // B * 9801

    {
        int total4 = B * PLANE4;
        sumsq_kernel<<<(total4 + 255) / 256, 256, 0, stream>>>(feat, t, total4);
    }
    {
        int total = B * RPLP;
        rowsum_kernel<<<(total + 255) / 256, 256, 0, stream>>>(t, r, total);
    }
    {
        int blocks = B * 7 * 7;           // one wave per 16x16 output tile
        colpool_wmma<<<blocks, 32, 0, stream>>>(r, pooled);
    }
    {
        int total = B * PPL;
        neigh_kernel<<<(total + 255) / 256, 256, 0, stream>>>(pooled, out, total);
    }
}